// DisambiguationGCNAE_42056319762467
// MI455X (gfx1250) — compile-verified
//
#include <hip/hip_runtime.h>

typedef float v2f __attribute__((ext_vector_type(2)));
typedef float v8f __attribute__((ext_vector_type(8)));

#define N_NODES 8192

// ---------------------------------------------------------------------------
// Zero-fill scratch (float4 stores)
// ---------------------------------------------------------------------------
__global__ void zero_f32_kernel(float4* __restrict__ p, int n4) {
    int i = blockIdx.x * blockDim.x + threadIdx.x;
    if (i < n4) p[i] = make_float4(0.f, 0.f, 0.f, 0.f);
}

// ---------------------------------------------------------------------------
// FP32 WMMA GEMM: C[M x N] = op(A) @ op(B), full fp32 via V_WMMA_F32_16X16X4_F32.
//   RELU_A : apply relu to A elements on load (fuses the GCN activation)
//   B_NT   : B is stored as N x K row-major (computes A @ B^T, used for Z@Z^T)
// One wave (32 lanes) owns one 16x16 C tile; K loop steps by 4.
//
// VGPR layouts per CDNA5 ISA 7.12.2 (16x16x4 f32):
//   A (16x4):  lanes 0-15 -> M=lane, {v0,v1}={K0,K1}; lanes 16-31 -> {K2,K3}
//   B (4x16):  lanes 0-15 -> N=lane, {v0,v1}={K0,K1}; lanes 16-31 -> {K2,K3}
//   C/D 16x16: vgpr r: lanes 0-15 -> M=r,N=lane; lanes 16-31 -> M=r+8,N=lane-16
// ---------------------------------------------------------------------------
template <bool RELU_A, bool B_NT>
__launch_bounds__(256)
__global__ void wmma_gemm_f32_kernel(const float* __restrict__ A,
                                     const float* __restrict__ B,
                                     float* __restrict__ C,
                                     int M, int N, int K,
                                     int lda, int ldb, int ldc) {
    const int lane = threadIdx.x;            // 0..31
    const int half = lane >> 4;              // 0 or 1
    const int lrow = lane & 15;              // 0..15
    const int tileN = blockIdx.x * blockDim.y + threadIdx.y;
    const int tileM = blockIdx.y;
    const int m0 = tileM * 16;
    const int n0 = tileN * 16;
    if (n0 >= N || m0 >= M) return;          // uniform per wave

    v8f acc = {0.f, 0.f, 0.f, 0.f, 0.f, 0.f, 0.f, 0.f};

    const float* Arow = A + (size_t)(m0 + lrow) * lda;
    const float* Brow = B_NT ? (B + (size_t)(n0 + lrow) * ldb) : nullptr;

    for (int k0 = 0; k0 < K; k0 += 4) {
        const int ka = k0 + 2 * half;        // this half-wave's K pair
        v2f a;
        a.x = Arow[ka];
        a.y = Arow[ka + 1];
        if (RELU_A) {
            a.x = fmaxf(a.x, 0.f);
            a.y = fmaxf(a.y, 0.f);
        }
        v2f b;
        if (B_NT) {
            b.x = Brow[ka];
            b.y = Brow[ka + 1];
        } else {
            b.x = B[(size_t)ka * ldb + (n0 + lrow)];
            b.y = B[(size_t)(ka + 1) * ldb + (n0 + lrow)];
        }
        // 8 args: (neg_a, A, neg_b, B, c_mod, C, reuse_a, reuse_b)
        acc = __builtin_amdgcn_wmma_f32_16x16x4_f32(
            false, a, false, b, (short)0, acc, false, false);
    }

#pragma unroll
    for (int r = 0; r < 8; ++r) {
        C[(size_t)(m0 + r + 8 * half) * ldc + (n0 + lrow)] = acc[r];
    }
}

// ---------------------------------------------------------------------------
// SpMM scatter: out[dst[e]*F + f] += w[e] * h[src[e]*F + f]
// One thread per (edge, feature). h and out are L2-resident (<= 8 MiB).
// ---------------------------------------------------------------------------
template <int LOGF>
__launch_bounds__(256)
__global__ void spmm_scatter_kernel(const int* __restrict__ esrc,
                                    const int* __restrict__ edst,
                                    const float* __restrict__ ew,
                                    const float* __restrict__ h,
                                    float* __restrict__ out,
                                    long long total) {
    long long idx = (long long)blockIdx.x * blockDim.x + threadIdx.x;
    if (idx >= total) return;
    const int F = 1 << LOGF;
    int e = (int)(idx >> LOGF);
    int f = (int)(idx & (F - 1));
    float val = ew[e] * h[((size_t)esrc[e] << LOGF) + f];
    atomicAdd(&out[((size_t)edst[e] << LOGF) + f], val);
}

// ---------------------------------------------------------------------------
// Orchestration
// Workspace layout (floats):
//   H    = ws + 0                 : 8192*256  (X @ W1)
//   AGG1 = H    + 8192*256        : 8192*256  (spmm of H)
//   Z0   = AGG1 + 8192*256        : 8192*64   (relu(AGG1) @ W2)
//   Z    = Z0   + 8192*64         : 8192*64   (spmm of Z0)
// Total: 8192*640 floats = 20 MiB
// ---------------------------------------------------------------------------
extern "C" void kernel_launch(void* const* d_in, const int* in_sizes, int n_in,
                              void* d_out, int out_size, void* d_ws, size_t ws_size,
                              hipStream_t stream) {
    const float* x    = (const float*)d_in[0];
    const int*   esrc = (const int*)d_in[1];
    const int*   edst = (const int*)d_in[2];
    const float* ew   = (const float*)d_in[3];
    const float* W1   = (const float*)d_in[4];
    const float* W2   = (const float*)d_in[5];
    float*       out  = (float*)d_out;
    const int nE = in_sizes[1];

    float* H    = (float*)d_ws;
    float* AGG1 = H + (size_t)N_NODES * 256;
    float* Z0   = AGG1 + (size_t)N_NODES * 256;
    float* Z    = Z0 + (size_t)N_NODES * 64;

    // Zero AGG1..Z (contiguous span: 8192*(256+64+64) floats; Z0 is
    // fully overwritten by GEMM2 anyway, so blanket-zero is fine).
    {
        int n4 = (N_NODES * (256 + 64 + 64)) / 4;
        zero_f32_kernel<<<(n4 + 255) / 256, 256, 0, stream>>>((float4*)AGG1, n4);
    }

    // GEMM1: H = X @ W1   (8192x512)@(512x256)
    wmma_gemm_f32_kernel<false, false>
        <<<dim3(2, N_NODES / 16), dim3(32, 8), 0, stream>>>(
            x, W1, H, N_NODES, 256, 512, 512, 256, 256);

    // SpMM1: AGG1[dst] += w * H[src]   (F = 256)
    {
        long long total = (long long)nE << 8;
        spmm_scatter_kernel<8>
            <<<(unsigned)((total + 255) / 256), 256, 0, stream>>>(
                esrc, edst, ew, H, AGG1, total);
    }

    // GEMM2: Z0 = relu(AGG1) @ W2   (8192x256)@(256x64)
    wmma_gemm_f32_kernel<true, false>
        <<<dim3(1, N_NODES / 16), dim3(32, 4), 0, stream>>>(
            AGG1, W2, Z0, N_NODES, 64, 256, 256, 64, 64);

    // SpMM2: Z[dst] += w * Z0[src]   (F = 64)
    {
        long long total = (long long)nE << 6;
        spmm_scatter_kernel<6>
            <<<(unsigned)((total + 255) / 256), 256, 0, stream>>>(
                esrc, edst, ew, Z0, Z, total);
    }

    // GEMM3: out = Z @ Z^T   (8192x64)@(64x8192) -> 8192x8192
    wmma_gemm_f32_kernel<false, true>
        <<<dim3(64, N_NODES / 16), dim3(32, 8), 0, stream>>>(
            Z, Z, out, N_NODES, N_NODES, 64, 64, 64, N_NODES);
}